// Range_Fourier_Net_18811956756731
// MI455X (gfx1250) — compile-verified
//
#include <hip/hip_runtime.h>

typedef __attribute__((ext_vector_type(2))) float v2f;
typedef __attribute__((ext_vector_type(4))) float v4f;
typedef __attribute__((ext_vector_type(8))) float v8f;
typedef __attribute__((ext_vector_type(4))) int   v4i;

#define N_DFT    512
#define BM        32                 // rows per workgroup tile
#define BN        64                 // cols per workgroup tile
#define BK        32                 // K chunk staged in LDS
#define LDSTRIDE  36                 // 144B rows: 16B aligned, bank-conflict-free
#define NCHUNK   (N_DFT / BK)        // 16

#define XTILE (BM * LDSTRIDE)        // floats per X (re or im) tile
#define WTILE (BN * LDSTRIDE)        // floats per W (re or im) tile

#if __has_builtin(__builtin_amdgcn_global_load_async_to_lds_b128)
#define USE_ASYNC 1
#else
#define USE_ASYNC 0
#endif

__device__ __forceinline__ void wait_async0() {
#if __has_builtin(__builtin_amdgcn_s_wait_asynccnt)
    __builtin_amdgcn_s_wait_asynccnt(0);
#else
    asm volatile("s_wait_asynccnt 0x0" ::: "memory");
#endif
}

#if USE_ASYNC
__device__ __forceinline__ void async_cp16(const float* g, float* l) {
    // GLOBAL_LOAD_ASYNC_TO_LDS_B128: 16B global -> LDS, tracked by ASYNCcnt
    __builtin_amdgcn_global_load_async_to_lds_b128(
        (__attribute__((address_space(1))) v4i*)(g),
        (__attribute__((address_space(3))) v4i*)(l),
        0, 0);
}
#endif

__global__ __launch_bounds__(256, 2)
void dft512_cplx_wmma_f32(const float* __restrict__ xre,
                          const float* __restrict__ xim,
                          const float* __restrict__ wre,
                          const float* __restrict__ wim,
                          float* __restrict__ out)
{
    __shared__ float sX[2][2 * XTILE];   // [buf][ re | im ]
    __shared__ float sW[2][2 * WTILE];   // [buf][ re | im ]

    const int tid   = threadIdx.x;
    const int lane  = tid & 31;
    const int wave  = tid >> 5;      // 0..7
    const int waveM = wave >> 2;     // 0..1
    const int waveN = wave & 3;      // 0..3
    const int hh    = lane >> 4;     // half-wave
    const int lm    = lane & 15;

    const int m0Blk = blockIdx.y * BM;
    const int n0Blk = blockIdx.x * BN;

    v8f acc_rr = {}; // Xre*Wr^T
    v8f acc_ii = {}; // Xim*Wi^T
    v8f acc_ri = {}; // Xre*Wi^T
    v8f acc_ir = {}; // Xim*Wr^T

    // cooperative staging indices (256 threads, one b128 each per tile pass)
    const int xr  = tid >> 3;            // 0..31
    const int xc  = (tid & 7) << 2;      // 0,4,..,28
    const int wr1 = (tid + 256) >> 3;    // 32..63

    const size_t gX  = (size_t)(m0Blk + xr ) * N_DFT + xc;
    const size_t gW0 = (size_t)(n0Blk + xr ) * N_DFT + xc;
    const size_t gW1 = (size_t)(n0Blk + wr1) * N_DFT + xc;
    const int lX  = xr  * LDSTRIDE + xc;
    const int lW0 = xr  * LDSTRIDE + xc;
    const int lW1 = wr1 * LDSTRIDE + xc;

    auto compute = [&](int buf) {
        const float* aRe = &sX[buf][(waveM * 16 + lm) * LDSTRIDE + 2 * hh];
        const float* aIm = aRe + XTILE;
        const float* bRe = &sW[buf][(waveN * 16 + lm) * LDSTRIDE + 2 * hh];
        const float* bIm = bRe + WTILE;
        #pragma unroll
        for (int kk = 0; kk < BK; kk += 4) {
            v2f ar = *(const v2f*)(aRe + kk);
            v2f ai = *(const v2f*)(aIm + kk);
            v2f br = *(const v2f*)(bRe + kk);
            v2f bi = *(const v2f*)(bIm + kk);
            acc_rr = __builtin_amdgcn_wmma_f32_16x16x4_f32(false, ar, false, br, (short)0, acc_rr, false, false);
            acc_ii = __builtin_amdgcn_wmma_f32_16x16x4_f32(false, ai, false, bi, (short)0, acc_ii, false, false);
            acc_ri = __builtin_amdgcn_wmma_f32_16x16x4_f32(false, ar, false, bi, (short)0, acc_ri, false, false);
            acc_ir = __builtin_amdgcn_wmma_f32_16x16x4_f32(false, ai, false, br, (short)0, acc_ir, false, false);
        }
    };

#if USE_ASYNC
    // ---- async double-buffered pipeline: one barrier per chunk ----
    auto stage_async = [&](int buf, int k0) {
        async_cp16(xre + gX  + k0, &sX[buf][lX]);
        async_cp16(xim + gX  + k0, &sX[buf][XTILE + lX]);
        async_cp16(wre + gW0 + k0, &sW[buf][lW0]);
        async_cp16(wre + gW1 + k0, &sW[buf][lW1]);
        async_cp16(wim + gW0 + k0, &sW[buf][WTILE + lW0]);
        async_cp16(wim + gW1 + k0, &sW[buf][WTILE + lW1]);
    };

    stage_async(0, 0);
    for (int ko = 0; ko < NCHUNK; ++ko) {
        const int cur = ko & 1;
        wait_async0();        // copies issued last iteration are done (hidden behind WMMAs)
        __syncthreads();      // publish buf[cur] to all waves
        if (ko + 1 < NCHUNK)
            stage_async(cur ^ 1, (ko + 1) * BK);   // overlap with compute below
        compute(cur);
    }
#else
    // ---- register-staged double-buffered pipeline ----
    v4f vxr, vxi, vw0r, vw1r, vw0i, vw1i;
    auto load_regs = [&](int k0) {
        vxr  = *(const v4f*)(xre + gX  + k0);
        vxi  = *(const v4f*)(xim + gX  + k0);
        vw0r = *(const v4f*)(wre + gW0 + k0);
        vw1r = *(const v4f*)(wre + gW1 + k0);
        vw0i = *(const v4f*)(wim + gW0 + k0);
        vw1i = *(const v4f*)(wim + gW1 + k0);
    };
    auto store_lds = [&](int buf) {
        *(v4f*)(&sX[buf][lX])          = vxr;
        *(v4f*)(&sX[buf][XTILE + lX])  = vxi;
        *(v4f*)(&sW[buf][lW0])         = vw0r;
        *(v4f*)(&sW[buf][lW1])         = vw1r;
        *(v4f*)(&sW[buf][WTILE + lW0]) = vw0i;
        *(v4f*)(&sW[buf][WTILE + lW1]) = vw1i;
    };

    load_regs(0);
    store_lds(0);
    __syncthreads();
    for (int ko = 0; ko < NCHUNK; ++ko) {
        const int cur = ko & 1;
        if (ko + 1 < NCHUNK) load_regs((ko + 1) * BK);  // issue early, hide latency
        compute(cur);
        if (ko + 1 < NCHUNK) store_lds(cur ^ 1);        // loadcnt wait lands here
        __syncthreads();
    }
#endif

    // fused complex epilogue + interleaved {re,im} store
    const int mBase = m0Blk + waveM * 16 + 8 * hh;
    const int n     = n0Blk + waveN * 16 + lm;
    #pragma unroll
    for (int v = 0; v < 8; ++v) {
        v2f y;
        y.x = acc_rr[v] - acc_ii[v];
        y.y = acc_ri[v] + acc_ir[v];
        *(v2f*)(out + ((size_t)(mBase + v) * N_DFT + n) * 2) = y;
    }
}

extern "C" void kernel_launch(void* const* d_in, const int* in_sizes, int n_in,
                              void* d_out, int out_size, void* d_ws, size_t ws_size,
                              hipStream_t stream) {
    const float* xre = (const float*)d_in[0];
    const float* xim = (const float*)d_in[1];
    const float* wre = (const float*)d_in[2];
    const float* wim = (const float*)d_in[3];
    float* out = (float*)d_out;

    const int M = in_sizes[0] / N_DFT;      // 8*16*256 = 32768
    dim3 grid(N_DFT / BN, M / BM);          // (8, 1024)
    dim3 block(256);
    hipLaunchKernelGGL(dft512_cplx_wmma_f32, grid, block, 0, stream,
                       xre, xim, wre, wim, out);
}